// FPSGrouping_40561671143694
// MI455X (gfx1250) — compile-verified
//
#include <hip/hip_runtime.h>
#include <stdint.h>

// Problem constants (match reference)
#define B_       32
#define N_       131072
#define NPTS     2048
#define SLICES   8                       // blocks per batch
#define SLICE_N  (N_ / SLICES)           // 16384 points per block
#define THREADS  1024                    // 32 wave32s = one full WGP
#define PPT      (SLICE_N / THREADS)     // 16 points per thread (registers)
#define INIT_DIST 1e10f

#if defined(__has_builtin)
#  if __has_builtin(__builtin_amdgcn_tensor_load_to_lds) && __has_builtin(__builtin_amdgcn_s_wait_tensorcnt)
#    define HAVE_TDM 1
#  endif
#endif

typedef unsigned int tdm_u32x4 __attribute__((ext_vector_type(4)));
typedef int          tdm_i32x8 __attribute__((ext_vector_type(8)));
typedef int          tdm_i32x4 __attribute__((ext_vector_type(4)));

// 8-way cross-block candidate exchange slot (L2-resident, parity double-buffered)
struct Cand {
    unsigned long long key;   // (dist_bits << 32) | ~global_idx
    float x, y, z;
    unsigned pad;
};

__global__ void fps_zero_cnt(unsigned* cnt, int n) {
    int i = blockIdx.x * blockDim.x + threadIdx.x;
    if (i < n) cnt[i] = 0u;
}

__global__ __launch_bounds__(THREADS, 1)
void fps_main(const float* __restrict__ xyz, float* __restrict__ out,
              unsigned* __restrict__ cnt, Cand* __restrict__ cand) {
    const int b    = blockIdx.x >> 3;        // batch
    const int s    = blockIdx.x & 7;         // slice within batch
    const int tid  = threadIdx.x;
    const int base = s * SLICE_N;            // first global point index of slice

    __shared__ float sxyz[SLICE_N * 3];      // 192 KB point slice
    __shared__ unsigned long long sKey;      // block-level argmax key
    __shared__ float sc[3];                  // current centroid broadcast

    const float* gsrc = xyz + ((size_t)b * N_ + base) * 3;

    // ---- Stage slice into LDS via Tensor Data Mover (one DMA per block) ----
#if defined(HAVE_TDM)
    if ((tid >> 5) == 0) {  // wave 0 only: TDM is a wave-level DMA issue
        const unsigned elems  = SLICE_N * 3;                 // 49152 dwords
        unsigned lds_off      = (unsigned)(uintptr_t)(&sxyz[0]); // low 32 bits of generic ptr = LDS byte offset
        unsigned long long ga = (unsigned long long)(uintptr_t)gsrc;

        tdm_u32x4 g0;
        g0.x = 1u;                                           // count=1, user descriptor
        g0.y = lds_off;                                      // lds_addr
        g0.z = (unsigned)ga;                                 // global_addr[31:0]
        g0.w = (unsigned)((ga >> 32) & 0x1FFFFFFu) | (2u << 30); // global_addr[56:32] | type=2

        tdm_i32x8 g1;
        g1[0] = (int)(2u << 16);                             // data_size=4B, mask=0, no flags
        g1[1] = (int)((elems & 0xFFFFu) << 16);              // tensor_dim0[15:0] @ bits 63:48
        g1[2] = (int)(((elems >> 16) & 0xFFFFu) | (1u << 16)); // tensor_dim0[31:16] | tensor_dim1[15:0]=1
        g1[3] = (int)((elems & 0xFFFFu) << 16);              // tensor_dim1[31:16]=0 | tile_dim0=elems (0xC000)
        g1[4] = 1;                                           // tile_dim1=1, tile_dim2=0
        g1[5] = (int)elems;                                  // tensor_dim0_stride[31:0]
        g1[6] = 0;                                           // stride hi / dim1_stride lo
        g1[7] = 0;
        tdm_i32x4 z4 = {0, 0, 0, 0};                         // groups 2/3 unused (<=2D)
        tdm_i32x8 z8 = {0, 0, 0, 0, 0, 0, 0, 0};             // extra group slot (6-arg form)
        __builtin_amdgcn_tensor_load_to_lds(g0, g1, z4, z4, z8, 0);
        __builtin_amdgcn_s_wait_tensorcnt(0);
    }
#else
    for (int i = tid; i < SLICE_N * 3; i += THREADS) sxyz[i] = gsrc[i];
#endif
    __syncthreads();

    // ---- Cache my 16 points + running min-distances in VGPRs ----
    float px[PPT], py[PPT], pz[PPT], dist[PPT];
#pragma unroll
    for (int k = 0; k < PPT; ++k) {
        int i = k * THREADS + tid;
        px[k] = sxyz[i * 3 + 0];
        py[k] = sxyz[i * 3 + 1];
        pz[k] = sxyz[i * 3 + 2];
        dist[k] = INIT_DIST;
    }
    if (tid == 0) {  // first centroid is deterministic index 0 of the batch
        const float* p0 = xyz + (size_t)b * N_ * 3;
        sc[0] = p0[0]; sc[1] = p0[1]; sc[2] = p0[2];
    }
    __syncthreads();

    unsigned* myCnt = cnt + (size_t)b * NPTS;

    for (int it = 0; it < NPTS; ++it) {
        const float ccx = sc[0], ccy = sc[1], ccz = sc[2];
        if (s == 0 && tid == 0) {            // emit current farthest's coords
            float* o = out + ((size_t)b * NPTS + it) * 3;
            o[0] = ccx; o[1] = ccy; o[2] = ccz;
        }
        if (tid == 0) sKey = 0ull;

        // min-distance update + thread-local argmax (all in registers)
        float bd = -1.0f, bx = 0.f, by = 0.f, bz = 0.f;
        unsigned bi = 0u;
#pragma unroll
        for (int k = 0; k < PPT; ++k) {
            float dx = px[k] - ccx;
            float dy = py[k] - ccy;
            float dz = pz[k] - ccz;
            float d  = fmaf(dz, dz, fmaf(dy, dy, dx * dx));
            float nd = fminf(dist[k], d);
            dist[k]  = nd;
            if (nd > bd) {                   // strict > keeps smallest index on ties
                bd = nd; bx = px[k]; by = py[k]; bz = pz[k];
                bi = (unsigned)(base + k * THREADS + tid);
            }
        }
        // key: larger dist wins; ties -> smaller global index (matches jnp.argmax)
        const unsigned long long my =
            ((unsigned long long)__float_as_uint(bd) << 32) | (unsigned)(~bi);

        __syncthreads();                     // sKey reset visible; sc reads done

        // wave32 reduction, then one ds_max_u64 per wave
        unsigned long long wk = my;
#pragma unroll
        for (int off = 16; off > 0; off >>= 1) {
            unsigned long long o = __shfl_xor(wk, off, 32);
            wk = (o > wk) ? o : wk;
        }
        if ((tid & 31) == 0) atomicMax(&sKey, wk);
        __syncthreads();

        // unique block winner publishes candidate to L2 slot
        Cand* slot = cand + ((size_t)(b * 2 + (it & 1)) * SLICES);
        if (my == sKey) {
            Cand* c = slot + s;
            c->key = my; c->x = bx; c->y = by; c->z = bz;
            __threadfence();                 // release winner's store device-wide
        }
        __syncthreads();                     // winner fence ordered before arrive

        if (tid == 0) {
            atomicAdd(&myCnt[it], 1u);       // arrive (monotonic counter, no reset races)
            while (atomicAdd(&myCnt[it], 0u) < SLICES) {
                __builtin_amdgcn_s_sleep(2);
            }
            __threadfence();                 // acquire before reading peers' candidates
            volatile Cand* cc = (volatile Cand*)slot;
            unsigned long long bb = 0ull;
            float wx = 0.f, wy = 0.f, wz = 0.f;
            for (int j = 0; j < SLICES; ++j) {
                unsigned long long kj = cc[j].key;
                if (kj > bb) { bb = kj; wx = cc[j].x; wy = cc[j].y; wz = cc[j].z; }
            }
            sc[0] = wx; sc[1] = wy; sc[2] = wz;   // next centroid
        }
        __syncthreads();
    }
}

extern "C" void kernel_launch(void* const* d_in, const int* in_sizes, int n_in,
                              void* d_out, int out_size, void* d_ws, size_t ws_size,
                              hipStream_t stream) {
    (void)in_sizes; (void)n_in; (void)out_size; (void)ws_size;
    const float* points = (const float*)d_in[0];
    float* out = (float*)d_out;

    // ws layout: [ cnt: B*NPTS u32 (256 KB) ][ cand: B*2*SLICES Cand slots ]
    unsigned* cnt = (unsigned*)d_ws;
    Cand* cand    = (Cand*)((char*)d_ws + (size_t)B_ * NPTS * sizeof(unsigned));

    const int ncnt = B_ * NPTS;
    fps_zero_cnt<<<(ncnt + 255) / 256, 256, 0, stream>>>(cnt, ncnt);
    fps_main<<<B_ * SLICES, THREADS, 0, stream>>>(points, out, cnt, cand);
}